// supervised_38439957299961
// MI455X (gfx1250) — compile-verified
//
#include <hip/hip_runtime.h>

// Problem constants (match reference)
#define NNODES 100000
#define NEDGES 1600000
#define HDIM   128
#define CDIM   64
#define NGRAPH 1000
#define NQRY   512

typedef __attribute__((ext_vector_type(2))) float v2f;
typedef __attribute__((ext_vector_type(8))) float v8f;

// ---------------- small utility kernels ----------------

__global__ void fill_f32(float* __restrict__ p, float v, int n) {
    int t = blockIdx.x * blockDim.x + threadIdx.x;
    if (t < n) p[t] = v;
}

__global__ void deg_edges(const int* __restrict__ dst, float* __restrict__ deg) {
    int t = blockIdx.x * blockDim.x + threadIdx.x;
    if (t < NEDGES) unsafeAtomicAdd(&deg[dst[t]], 1.0f);
}

__global__ void rsqrt_inplace(float* __restrict__ d) {
    int t = blockIdx.x * blockDim.x + threadIdx.x;
    if (t < NNODES) d[t] = rsqrtf(d[t]);
}

__global__ void edge_coef(const int* __restrict__ src, const int* __restrict__ dst,
                          const float* __restrict__ dinv, float* __restrict__ coef) {
    int t = blockIdx.x * blockDim.x + threadIdx.x;
    if (t < NEDGES) coef[t] = dinv[src[t]] * dinv[dst[t]];
}

// ---------------- WMMA fp32 GEMM: out[N,16*NT] = X[N,128] @ W[128,16*NT] ----------------
// block = 128 threads = 4 waves, each wave computes a 16-row strip x (16*NT) cols.
// A layout (32-bit, 16x4): lanes 0-15 hold M=lane, K=k0+{0,1}; lanes 16-31 K=k0+{2,3}.
// B layout mirrored; C/D: VGPR r -> M=r (lanes 0-15) / M=r+8 (lanes 16-31), N=lane&15.
//
// W is staged in LDS pair-interleaved: sW[p*NC + n] = {W[2p][n], W[2p+1][n]} so each
// B fragment is one aligned ds_load_b64 straight into an even VGPR pair (no repack movs).
template <int NT>
__global__ __launch_bounds__(128) void gemm_tiles(const float* __restrict__ X,
                                                  const float* __restrict__ W,
                                                  float* __restrict__ out,
                                                  int nstrips) {
    constexpr int NC = NT * 16;
    __shared__ v2f sW[(HDIM / 2) * NC];
    for (int i = threadIdx.x; i < (HDIM / 2) * NC; i += 128) {
        int p = i / NC;
        int n = i % NC;
        v2f v;
        v.x = W[(size_t)(2 * p) * NC + n];
        v.y = W[(size_t)(2 * p + 1) * NC + n];
        sW[i] = v;
    }
    __syncthreads();

    const int lane = threadIdx.x & 31;
    const int wave = threadIdx.x >> 5;
    const int strip = blockIdx.x * 4 + wave;
    if (strip >= nstrips) return;  // whole-wave uniform: EXEC stays all-ones below

    const int mrow = strip * 16 + (lane & 15);
    const int koff = (lane < 16) ? 0 : 2;   // A/B fragment K sub-offset for this lane half
    const int prow = koff >> 1;             // pair-row offset into interleaved LDS
    const int ncol = lane & 15;
    const float* xrow = X + (size_t)mrow * HDIM;

    v8f acc[NT] = {};
#pragma unroll 4
    for (int k0 = 0; k0 < HDIM; k0 += 4) {
        v2f a = *(const v2f*)(xrow + k0 + koff);  // contiguous global b64 load
        const v2f* brow = &sW[((k0 >> 1) + prow) * NC + ncol];
#pragma unroll
        for (int t = 0; t < NT; ++t) {
            v2f b = brow[t * 16];  // single aligned ds_load_b64
            acc[t] = __builtin_amdgcn_wmma_f32_16x16x4_f32(
                false, a, false, b, (short)0, acc[t], false, false);
        }
    }

    const int rbase = strip * 16 + ((lane < 16) ? 0 : 8);
#pragma unroll
    for (int t = 0; t < NT; ++t)
#pragma unroll
        for (int r = 0; r < 8; ++r)
            out[(size_t)(rbase + r) * NC + t * 16 + ncol] = acc[t][r];
}

// Final head GEMM: raw = X@Wm + bm, tile results atomically pooled into sums[batch[row]].
__global__ __launch_bounds__(128) void gemm_pool(const float* __restrict__ X,
                                                 const float* __restrict__ Wm,
                                                 const float* __restrict__ bm,
                                                 const int* __restrict__ batch,
                                                 float* __restrict__ sums,
                                                 int nstrips) {
    constexpr int NT = CDIM / 16;  // 4
    constexpr int NC = CDIM;
    __shared__ v2f sW[(HDIM / 2) * NC];
    for (int i = threadIdx.x; i < (HDIM / 2) * NC; i += 128) {
        int p = i / NC;
        int n = i % NC;
        v2f v;
        v.x = Wm[(size_t)(2 * p) * NC + n];
        v.y = Wm[(size_t)(2 * p + 1) * NC + n];
        sW[i] = v;
    }
    __syncthreads();

    const int lane = threadIdx.x & 31;
    const int wave = threadIdx.x >> 5;
    const int strip = blockIdx.x * 4 + wave;
    if (strip >= nstrips) return;

    const int mrow = strip * 16 + (lane & 15);
    const int koff = (lane < 16) ? 0 : 2;
    const int prow = koff >> 1;
    const int ncol = lane & 15;
    const float* xrow = X + (size_t)mrow * HDIM;

    v8f acc[NT] = {};
#pragma unroll 4
    for (int k0 = 0; k0 < HDIM; k0 += 4) {
        v2f a = *(const v2f*)(xrow + k0 + koff);
        const v2f* brow = &sW[((k0 >> 1) + prow) * NC + ncol];
#pragma unroll
        for (int t = 0; t < NT; ++t) {
            v2f b = brow[t * 16];
            acc[t] = __builtin_amdgcn_wmma_f32_16x16x4_f32(
                false, a, false, b, (short)0, acc[t], false, false);
        }
    }

    const int rbase = strip * 16 + ((lane < 16) ? 0 : 8);
#pragma unroll
    for (int t = 0; t < NT; ++t) {
        float bias = bm[t * 16 + ncol];
#pragma unroll
        for (int r = 0; r < 8; ++r) {
            int row = rbase + r;
            int g = batch[row];
            unsafeAtomicAdd(&sums[(size_t)g * CDIM + t * 16 + ncol], acc[t][r] + bias);
        }
    }
}

// ---------------- per-node init: out = xw*dinv^2 + bias (+ residual) ----------------
__global__ void init_agg(const float* __restrict__ xw, const float* __restrict__ dinv,
                         const float* __restrict__ bias, const float* __restrict__ res,
                         float* __restrict__ out) {
    int t = blockIdx.x * blockDim.x + threadIdx.x;  // NNODES*32 threads, float4 each
    if (t >= NNODES * 32) return;
    int n = t >> 5;
    int lane = t & 31;
    float c = dinv[n];
    c = c * c;  // self-loop coefficient
    float4 v = ((const float4*)xw)[t];
    float4 b = ((const float4*)bias)[lane];
    float4 o;
    o.x = v.x * c + b.x;
    o.y = v.y * c + b.y;
    o.z = v.z * c + b.z;
    o.w = v.w * c + b.w;
    if (res != nullptr) {
        float4 r = ((const float4*)res)[t];
        o.x += r.x; o.y += r.y; o.z += r.z; o.w += r.w;
    }
    ((float4*)out)[t] = o;
}

// ---------------- edge scatter: out[dst] += xw[src]*coef ----------------
// 1 wave per edge: 32 lanes x float4 = 128 features, coalesced L2-resident gather.
__global__ void scatter_edges(const float* __restrict__ xw, const int* __restrict__ src,
                              const int* __restrict__ dst, const float* __restrict__ coef,
                              float* __restrict__ out) {
    int t = blockIdx.x * blockDim.x + threadIdx.x;  // NEDGES*32 threads
    int e = t >> 5;
    if (e >= NEDGES) return;
    int lane = t & 31;
    float c = coef[e];
    float4 v = ((const float4*)(xw + (size_t)src[e] * HDIM))[lane];
    float* o = out + (size_t)dst[e] * HDIM + lane * 4;
    unsafeAtomicAdd(o + 0, v.x * c);
    unsafeAtomicAdd(o + 1, v.y * c);
    unsafeAtomicAdd(o + 2, v.z * c);
    unsafeAtomicAdd(o + 3, v.w * c);
}

// ---------------- pooling count + final gather ----------------
__global__ void graph_cnt(const int* __restrict__ batch, float* __restrict__ cnt) {
    int t = blockIdx.x * blockDim.x + threadIdx.x;
    if (t < NNODES) unsafeAtomicAdd(&cnt[batch[t]], 1.0f);
}

__global__ void gather_out(const float* __restrict__ sums, const float* __restrict__ cnt,
                           const int* __restrict__ idx, float* __restrict__ out) {
    int t = blockIdx.x * blockDim.x + threadIdx.x;
    if (t >= NQRY * CDIM) return;
    int q = t / CDIM;
    int c = t % CDIM;
    int g = idx[q];
    out[t] = sums[(size_t)g * CDIM + c] / fmaxf(cnt[g], 1.0f);
}

// ---------------- host-side orchestration ----------------

extern "C" void kernel_launch(void* const* d_in, const int* in_sizes, int n_in,
                              void* d_out, int out_size, void* d_ws, size_t ws_size,
                              hipStream_t stream) {
    const float* x     = (const float*)d_in[0];
    const int*   ei    = (const int*)d_in[1];   // [2, E]
    const int*   idx   = (const int*)d_in[2];   // [NQ]
    const int*   batch = (const int*)d_in[3];   // [N]
    const float* W0    = (const float*)d_in[4];
    const float* b0    = (const float*)d_in[5];
    const float* W1    = (const float*)d_in[6];
    const float* b1    = (const float*)d_in[7];
    const float* W2    = (const float*)d_in[8];
    const float* b2    = (const float*)d_in[9];
    const float* Wm    = (const float*)d_in[10];
    const float* bm    = (const float*)d_in[11];
    const int* src = ei;
    const int* dst = ei + NEDGES;
    float* out = (float*)d_out;

    // workspace carve-out (floats)
    float* ws   = (float*)d_ws;
    float* xw   = ws;                       // N*128
    float* hA   = xw + (size_t)NNODES * HDIM;
    float* hB   = hA + (size_t)NNODES * HDIM;
    float* dinv = hB + (size_t)NNODES * HDIM;  // N (deg -> dinv in place)
    float* coef = dinv + NNODES;               // E
    float* sums = coef + NEDGES;               // G*C
    float* cnt  = sums + (size_t)NGRAPH * CDIM;  // G

    const int B = 256;
    const int nstrips = NNODES / 16;                 // 6250, exact
    const int gemm_blocks = (nstrips + 3) / 4;       // 4 waves/block
    dim3 gemm_block(128);

    // 1) normalization setup
    fill_f32<<<(NNODES + B - 1) / B, B, 0, stream>>>(dinv, 1.0f, NNODES);  // self-loop
    fill_f32<<<(NGRAPH * CDIM + NGRAPH + B - 1) / B, B, 0, stream>>>(sums, 0.0f,
                                                                     NGRAPH * CDIM + NGRAPH);
    deg_edges<<<(NEDGES + B - 1) / B, B, 0, stream>>>(dst, dinv);
    rsqrt_inplace<<<(NNODES + B - 1) / B, B, 0, stream>>>(dinv);
    edge_coef<<<(NEDGES + B - 1) / B, B, 0, stream>>>(src, dst, dinv, coef);

    const int initN = NNODES * 32;
    const int scatN = NEDGES * 32;

    // 2) layer 0: h0 = agg(x @ W0) + b0
    gemm_tiles<8><<<gemm_blocks, gemm_block, 0, stream>>>(x, W0, xw, nstrips);
    init_agg<<<(initN + B - 1) / B, B, 0, stream>>>(xw, dinv, b0, nullptr, hA);
    scatter_edges<<<(scatN + B - 1) / B, B, 0, stream>>>(xw, src, dst, coef, hA);

    // 3) layer 1: h1 = agg(h0 @ W1) + b1 + h0
    gemm_tiles<8><<<gemm_blocks, gemm_block, 0, stream>>>(hA, W1, xw, nstrips);
    init_agg<<<(initN + B - 1) / B, B, 0, stream>>>(xw, dinv, b1, hA, hB);
    scatter_edges<<<(scatN + B - 1) / B, B, 0, stream>>>(xw, src, dst, coef, hB);

    // 4) layer 2: h2 = agg(h1 @ W2) + b2 + h1
    gemm_tiles<8><<<gemm_blocks, gemm_block, 0, stream>>>(hB, W2, xw, nstrips);
    init_agg<<<(initN + B - 1) / B, B, 0, stream>>>(xw, dinv, b2, hB, hA);
    scatter_edges<<<(scatN + B - 1) / B, B, 0, stream>>>(xw, src, dst, coef, hA);

    // 5) head: sums[g] += (h2 @ Wm + bm) rows; counts; mean + gather
    graph_cnt<<<(NNODES + B - 1) / B, B, 0, stream>>>(batch, cnt);
    gemm_pool<<<gemm_blocks, gemm_block, 0, stream>>>(hA, Wm, bm, batch, sums, nstrips);
    gather_out<<<(NQRY * CDIM + B - 1) / B, B, 0, stream>>>(sums, cnt, idx, out);
}